// OrbitalSpectralConv_52321291600177
// MI455X (gfx1250) — compile-verified
//
#include <hip/hip_runtime.h>
#include <hip/hip_bf16.h>

// ---- problem constants (match reference) ----
#define Bn 8
#define Cn 128
#define Hn 128
#define Wn 128
#define MR 32          // kept kx rows
#define MCOL 17        // kept ky cols
#define NM (MR * MCOL) // 544 modes
#define STEP 0.049087385212340517f  // 2*pi/128

typedef __attribute__((ext_vector_type(2))) float v2f;
typedef __attribute__((ext_vector_type(8))) float v8f;

__device__ __forceinline__ v8f wmma_f32(v2f a, v2f b, v8f c) {
  // D = A(16x4) * B(4x16) + C(16x16), f32. 8-arg VOP3P form:
  // (neg_a, A, neg_b, B, c_mod, C, reuse_a, reuse_b); f32 has no A/B neg.
  return __builtin_amdgcn_wmma_f32_16x16x4_f32(false, a, false, b, (short)0, c,
                                               false, false);
}

// ---------------------------------------------------------------------------
// Kernel 0: zero-fill (Xf pad columns must be 0 for the WMMA B operand)
// ---------------------------------------------------------------------------
__global__ void k_zero(float* __restrict__ p, size_t n) {
  size_t i = (size_t)blockIdx.x * blockDim.x + threadIdx.x;
  size_t s = (size_t)gridDim.x * blockDim.x;
  for (; i < n; i += s) p[i] = 0.0f;
}

// ---------------------------------------------------------------------------
// Kernel 1: skew-Hermitian combine + transpose weights to mode-major
//   WT_re[m][t][c] = Lre[c,t,m] - Lre[t,c,m]
//   WT_im[m][t][c] = Lim[c,t,m] + Lim[t,c,m]
// grid = (C/16, C/16, NM/16), block = 256. Reads are 64B/thread contiguous
// (L2 absorbs the strided amplification); writes are 64B half-wave segments.
// ---------------------------------------------------------------------------
__global__ __launch_bounds__(256) void k_weff(const float* __restrict__ lre,
                                              const float* __restrict__ lim,
                                              float* __restrict__ wtr,
                                              float* __restrict__ wti) {
  const int tid = threadIdx.x;
  const int i = tid & 15;   // c-local (lanes 0..15 consecutive c -> coalesced)
  const int j = tid >> 4;   // t-local
  const int c = blockIdx.x * 16 + i;
  const int t = blockIdx.y * 16 + j;
  const int m0 = blockIdx.z * 16;

  const float* pT = lre + ((size_t)(c * Cn + t)) * NM + m0;
  const float* pU = lre + ((size_t)(t * Cn + c)) * NM + m0;
  float v[16];
#pragma unroll
  for (int q = 0; q < 4; ++q) {
    float4 a = *(const float4*)(pT + 4 * q);
    float4 b = *(const float4*)(pU + 4 * q);
    v[4 * q + 0] = a.x - b.x; v[4 * q + 1] = a.y - b.y;
    v[4 * q + 2] = a.z - b.z; v[4 * q + 3] = a.w - b.w;
  }
#pragma unroll
  for (int mm = 0; mm < 16; ++mm)
    wtr[((size_t)(m0 + mm) * Cn + t) * Cn + c] = v[mm];

  const float* qT = lim + ((size_t)(c * Cn + t)) * NM + m0;
  const float* qU = lim + ((size_t)(t * Cn + c)) * NM + m0;
#pragma unroll
  for (int q = 0; q < 4; ++q) {
    float4 a = *(const float4*)(qT + 4 * q);
    float4 b = *(const float4*)(qU + 4 * q);
    v[4 * q + 0] = a.x + b.x; v[4 * q + 1] = a.y + b.y;
    v[4 * q + 2] = a.z + b.z; v[4 * q + 3] = a.w + b.w;
  }
#pragma unroll
  for (int mm = 0; mm < 16; ++mm)
    wti[((size_t)(m0 + mm) * Cn + t) * Cn + c] = v[mm];
}

// ---------------------------------------------------------------------------
// Kernel 2: fused forward transforms, one block per (b,c).
//   Stage A (WMMA): Xw[h,ky] = (1/HW) * sum_w X[h,w] e^{-2pi i ky w/128}
//   Stage B (VALU): Xf[kx,ky] = sum_h Xw[h,ky] e^{-2pi i kx h/128}
// Output layout: Xf_{re,im}[m][c][16] with batch in cols 0..7 (8..15 zero).
// ---------------------------------------------------------------------------
__global__ __launch_bounds__(256) void k_fwd(const float* __restrict__ X,
                                             float* __restrict__ xfr,
                                             float* __restrict__ xfi) {
  __shared__ float xs[Hn][36];       // X panel: 128 h x 32 w  (stride 36 = 4*9)
  __shared__ float bsr[32][33];      // basis re: w-local x n(0..31)
  __shared__ float bsi[32][33];      // basis im
  __shared__ float xw[2][Hn][33];    // Xw re/im: h x n

  const int tid = threadIdx.x;
  const int bc = blockIdx.x;
  const int b = bc / Cn, c = bc % Cn;
  const float* Xp = X + (size_t)bc * Hn * Wn;
  const int lane15 = tid & 15;
  const int half = (tid >> 4) & 1;
  const int wave = tid >> 5;
  const float inv_hw = 1.0f / 16384.0f;

  v8f acc[4];
#pragma unroll
  for (int q = 0; q < 4; ++q) acc[q] = (v8f){};

  for (int p = 0; p < 4; ++p) {           // four 32-wide K panels
    const int w0 = p * 32;
    for (int idx = tid; idx < Hn * 32; idx += 256) {
      int h = idx >> 5, wl = idx & 31;
      xs[h][wl] = Xp[h * Wn + w0 + wl];
    }
    for (int idx = tid; idx < 32 * 32; idx += 256) {
      int wl = idx >> 5, n = idx & 31;
      float cr = 0.0f, ci = 0.0f;
      if (n < MCOL) {
        int ph = (n * (w0 + wl)) & 127;
        float s, co;
        __sincosf(ph * STEP, &s, &co);
        cr = co * inv_hw;                 // e^{-i theta}: re = cos
        ci = -s * inv_hw;                 //              im = -sin
      }
      bsr[wl][n] = cr;
      bsi[wl][n] = ci;
    }
    __syncthreads();

#pragma unroll
    for (int ci2 = 0; ci2 < 4; ++ci2) {
      const int combo = wave * 4 + ci2;   // (plane<<4)|(Mt<<1)|Nt
      const int Nt = combo & 1;
      const int Mt = (combo >> 1) & 7;
      const int plane = combo >> 4;
      float(*bp)[33] = plane ? bsi : bsr;
#pragma unroll
      for (int ks = 0; ks < 8; ++ks) {
        const int k0 = ks * 4 + 2 * half;
        v2f a, bb;
        a.x = xs[Mt * 16 + lane15][k0];
        a.y = xs[Mt * 16 + lane15][k0 + 1];
        bb.x = bp[k0][Nt * 16 + lane15];
        bb.y = bp[k0 + 1][Nt * 16 + lane15];
        acc[ci2] = wmma_f32(a, bb, acc[ci2]);
      }
    }
    __syncthreads();
  }

#pragma unroll
  for (int ci2 = 0; ci2 < 4; ++ci2) {
    const int combo = wave * 4 + ci2;
    const int Nt = combo & 1;
    const int Mt = (combo >> 1) & 7;
    const int plane = combo >> 4;
#pragma unroll
    for (int r = 0; r < 8; ++r)
      xw[plane][Mt * 16 + r + 8 * half][Nt * 16 + lane15] = acc[ci2][r];
  }
  __syncthreads();

  // Stage B: 544 outputs, each a 128-term complex reduction
  for (int idx = tid; idx < NM; idx += 256) {
    const int kxrow = idx / MCOL, ky = idx % MCOL;
    const int kx = (kxrow < 17) ? kxrow : kxrow + 96;  // rows 17..31 -> 113..127
    float re = 0.0f, im = 0.0f;
    for (int h = 0; h < Hn; ++h) {
      const int ph = (kx * h) & 127;
      float s, co;
      __sincosf(ph * STEP, &s, &co);
      const float xr = xw[0][h][ky], xi = xw[1][h][ky];
      re += xr * co + xi * s;     // (xr+i xi)*(co - i s)
      im += xi * co - xr * s;
    }
    xfr[((size_t)idx * Cn + c) * 16 + b] = re;
    xfi[((size_t)idx * Cn + c) * 16 + b] = im;
  }
}

// ---------------------------------------------------------------------------
// Kernel 3: per-mode complex channel matmul, Y = Weff * Xf   (WMMA)
//   M=c(128, 8 tiles = 8 waves), N=b(16, batch in 0..7), K=t(128, 32 steps)
// grid = NM, block = 256. A fragments read straight from mode-major WT.
// ---------------------------------------------------------------------------
__global__ __launch_bounds__(256) void k_mix(const float* __restrict__ wtr,
                                             const float* __restrict__ wti,
                                             const float* __restrict__ xfr,
                                             const float* __restrict__ xfi,
                                             float* __restrict__ yr,
                                             float* __restrict__ yi) {
  const int m = blockIdx.x;
  const int tid = threadIdx.x;
  const int lane15 = tid & 15;
  const int half = (tid >> 4) & 1;
  const int Mt = tid >> 5;

  const float* wr = wtr + (size_t)m * Cn * Cn;   // [t][c]
  const float* wi = wti + (size_t)m * Cn * Cn;
  const float* xr = xfr + (size_t)m * Cn * 16;   // [t][16]
  const float* xi = xfi + (size_t)m * Cn * 16;

  v8f accR = (v8f){}, accI = (v8f){};
  const int cI = Mt * 16 + lane15;

#pragma unroll 4
  for (int ks = 0; ks < 32; ++ks) {
    const int t0 = ks * 4 + 2 * half;
    v2f ar, ai, br, bi, bin;
    ar.x = wr[t0 * Cn + cI];        ar.y = wr[(t0 + 1) * Cn + cI];
    ai.x = wi[t0 * Cn + cI];        ai.y = wi[(t0 + 1) * Cn + cI];
    br.x = xr[t0 * 16 + lane15];    br.y = xr[(t0 + 1) * 16 + lane15];
    bi.x = xi[t0 * 16 + lane15];    bi.y = xi[(t0 + 1) * 16 + lane15];
    bin.x = -bi.x;                  bin.y = -bi.y;
    accR = wmma_f32(ar, br, accR);  // Yre += Wre*Xre
    accR = wmma_f32(ai, bin, accR); // Yre -= Wim*Xim
    accI = wmma_f32(ar, bi, accI);  // Yim += Wre*Xim
    accI = wmma_f32(ai, br, accI);  // Yim += Wim*Xre
  }
#pragma unroll
  for (int r = 0; r < 8; ++r) {
    const int cc = Mt * 16 + r + 8 * half;
    yr[((size_t)m * Cn + cc) * 16 + lane15] = accR[r];
    yi[((size_t)m * Cn + cc) * 16 + lane15] = accI[r];
  }
}

// ---------------------------------------------------------------------------
// Kernel 4: fused inverse transforms, one block per (b,c).
//   Stage C (VALU): Z[h,ky] = sum_kx Y[kx,ky] e^{+2pi i kx h/128}
//   Stage D (WMMA): out[h,w] = sum_ky f_ky (Zre cos - Zim sin), f_0=1 else 2
// Packed K=40: k 0..16 = Zre, k 20..36 = Zim (rest zero) -> single GEMM.
// ---------------------------------------------------------------------------
__global__ __launch_bounds__(256) void k_inv(const float* __restrict__ yr,
                                             const float* __restrict__ yi,
                                             float* __restrict__ out) {
  __shared__ float ysr[MR][18];
  __shared__ float ysi[MR][18];
  __shared__ float za[Hn][44];   // stride 44 = 4*11 -> conflict-free columns
  __shared__ float bd[40][132];  // basis: k x w

  const int tid = threadIdx.x;
  const int bc = blockIdx.x;
  const int b = bc / Cn, c = bc % Cn;

  for (int idx = tid; idx < NM; idx += 256) {
    const int kxrow = idx / MCOL, ky = idx % MCOL;
    ysr[kxrow][ky] = yr[((size_t)idx * Cn + c) * 16 + b];
    ysi[kxrow][ky] = yi[((size_t)idx * Cn + c) * 16 + b];
  }
  {
    float* zaf = &za[0][0];
    for (int idx = tid; idx < Hn * 44; idx += 256) zaf[idx] = 0.0f;
  }
  for (int idx = tid; idx < 40 * Wn; idx += 256) {
    const int k = idx >> 7, w = idx & 127;
    float v = 0.0f;
    if (k < MCOL) {
      const float f = (k == 0) ? 1.0f : 2.0f;
      const int ph = (k * w) & 127;
      v = f * __cosf(ph * STEP);
    } else if (k >= 20 && k < 20 + MCOL) {
      const int kk = k - 20;
      const float f = (kk == 0) ? 1.0f : 2.0f;
      const int ph = (kk * w) & 127;
      v = -f * __sinf(ph * STEP);
    }
    bd[k][w] = v;
  }
  __syncthreads();

  for (int idx = tid; idx < Hn * MCOL; idx += 256) {
    const int h = idx / MCOL, ky = idx % MCOL;
    float zr = 0.0f, zi = 0.0f;
    for (int r = 0; r < MR; ++r) {
      const int kx = (r < 17) ? r : r + 96;
      const int ph = (kx * h) & 127;
      float s, co;
      __sincosf(ph * STEP, &s, &co);
      const float vr = ysr[r][ky], vi = ysi[r][ky];
      zr += vr * co - vi * s;     // (vr+i vi)*(co + i s)
      zi += vr * s + vi * co;
    }
    za[h][ky] = zr;
    za[h][20 + ky] = zi;
  }
  __syncthreads();

  const int lane15 = tid & 15;
  const int half = (tid >> 4) & 1;
  const int wave = tid >> 5;
  float* outp = out + (size_t)bc * Hn * Wn;

#pragma unroll
  for (int tI = 0; tI < 8; ++tI) {
    const int tile = wave * 8 + tI;
    const int Mt = tile >> 3, Nt = tile & 7;
    v8f acc = (v8f){};
#pragma unroll
    for (int ks = 0; ks < 10; ++ks) {   // K = 40
      const int k0 = ks * 4 + 2 * half;
      v2f a, bb;
      a.x = za[Mt * 16 + lane15][k0];
      a.y = za[Mt * 16 + lane15][k0 + 1];
      bb.x = bd[k0][Nt * 16 + lane15];
      bb.y = bd[k0 + 1][Nt * 16 + lane15];
      acc = wmma_f32(a, bb, acc);
    }
#pragma unroll
    for (int r = 0; r < 8; ++r)
      outp[(Mt * 16 + r + 8 * half) * Wn + Nt * 16 + lane15] = acc[r];
  }
}

// ---------------------------------------------------------------------------
// launch
// ---------------------------------------------------------------------------
extern "C" void kernel_launch(void* const* d_in, const int* in_sizes, int n_in,
                              void* d_out, int out_size, void* d_ws,
                              size_t ws_size, hipStream_t stream) {
  (void)in_sizes; (void)n_in; (void)out_size; (void)ws_size;
  const float* X   = (const float*)d_in[0];
  const float* lre = (const float*)d_in[1];
  const float* lim = (const float*)d_in[2];

  float* ws  = (float*)d_ws;
  float* wtr = ws;                                   // NM*C*C
  float* wti = wtr + (size_t)NM * Cn * Cn;           // NM*C*C
  float* xfr = wti + (size_t)NM * Cn * Cn;           // NM*C*16
  float* xfi = xfr + (size_t)NM * Cn * 16;
  float* yr  = xfi + (size_t)NM * Cn * 16;
  float* yi  = yr  + (size_t)NM * Cn * 16;
  // total ws: 22,282,240 floats (~89 MB)

  k_zero<<<512, 256, 0, stream>>>(xfr, (size_t)NM * Cn * 16 * 2); // xfr+xfi
  dim3 gW(Cn / 16, Cn / 16, NM / 16);                             // 8,8,34
  k_weff<<<gW, 256, 0, stream>>>(lre, lim, wtr, wti);
  k_fwd<<<Bn * Cn, 256, 0, stream>>>(X, xfr, xfi);
  k_mix<<<NM, 256, 0, stream>>>(wtr, wti, xfr, xfi, yr, yi);
  k_inv<<<Bn * Cn, 256, 0, stream>>>(yr, yi, (float*)d_out);
}